// MSDeformableAttention3D_21947282882721
// MI455X (gfx1250) — compile-verified
//
#include <hip/hip_runtime.h>
#include <math.h>

typedef __attribute__((ext_vector_type(2))) float v2f;
typedef __attribute__((ext_vector_type(8))) float v8f;
typedef unsigned int u32;
typedef __attribute__((ext_vector_type(4))) u32 v4u;
typedef __attribute__((ext_vector_type(8))) u32 v8u;

#define BS 6
#define NQ 10000
#define EMB 256
#define NH 8
#define HD 32
#define NVAL 30825
#define MTOT (BS * NVAL)        // 184950
#define VPROJ_FLOATS 47347200   // 6*8*30825*32
#define WT_COLS 1024            // [w_off(512) | w_attn(256) | w_val(256)] transposed

__device__ __forceinline__ v8f wmma_f32(v2f a, v2f b, v8f c) {
  // V_WMMA_F32_16X16X4_F32 : D(16x16 f32) = A(16x4) * B(4x16) + C
  return __builtin_amdgcn_wmma_f32_16x16x4_f32(
      false, a, false, b, (short)0, c, false, false);
}

// ---------------------------------------------------------------------------
// Kernel 0: transpose [w_off | w_attn | w_val] -> wT[1024 cols][256 k]
// (contiguous-in-k B operands for kernel 2, TDM-friendly source for kernel 1)
// ---------------------------------------------------------------------------
__global__ void __launch_bounds__(256)
wt_transpose_kernel(const float* __restrict__ w_off, const float* __restrict__ w_attn,
                    const float* __restrict__ w_val, float* __restrict__ wT) {
  const int col = blockIdx.x * 256 + threadIdx.x;   // 0..1023
  const float* src; int Nw, n;
  if (col < 512)      { src = w_off;  Nw = 512; n = col; }
  else if (col < 768) { src = w_attn; Nw = 256; n = col - 512; }
  else                { src = w_val;  Nw = 256; n = col - 768; }
  float* dst = wT + (size_t)col * EMB;
#pragma unroll 8
  for (int k = 0; k < EMB; ++k) dst[k] = src[(size_t)k * Nw + n];
}

// ---------------------------------------------------------------------------
// Kernel 1: v[b,h,n,d] = value[b,n,:] @ w_val[:, h*32+d] + b_val
// Block = 256 threads (8 waves); 128 rows x 64 cols per block.
// B panel: TDM loads 64 transposed columns (256 k each) from wvT and inserts
// 4 DWORDs of padding every 256 DWORDs -> LDS [c][260], bank-conflict-free,
// with {B[k][c],B[k+1][c]} contiguous for direct ds_load_2addr operand pairs.
// ---------------------------------------------------------------------------
__global__ void __launch_bounds__(256)
val_proj_kernel(const float* __restrict__ value, const float* __restrict__ wvT,
                const float* __restrict__ b_val, float* __restrict__ vout) {
  extern __shared__ float ldsB[];   // [64 cols][260], 66560 B
  const int t = threadIdx.x;
  const int wave = t >> 5, lane = t & 31, hi = lane >> 4, lo = lane & 15;
  const int ncol0 = blockIdx.y * 64;

  if (t < 32) {  // wave 0 issues one TDM descriptor for the whole panel
    const unsigned long long ga =
        (unsigned long long)(const char*)(wvT + (size_t)ncol0 * EMB);
    v4u g0;
    g0.x = 1u;                            // count=1, user descriptor
    g0.y = 0u;                            // lds_addr = 0 (dynamic LDS base)
    g0.z = (u32)ga;                       // global_addr[31:0]
    g0.w = (u32)(ga >> 32) | (2u << 30);  // global_addr[56:32] | type=2
    v8u g1;
    g1[0] = (2u << 16)                    // data_size = 4 bytes
          | (1u << 20)                    // pad_enable
          | (7u << 22)                    // pad_interval: every 256 DWORDs
          | (3u << 25);                   // pad_amount: 4 DWORDs
    g1[1] = 256u << 16;                   // tensor_dim0 = 256 (bits 79:48)
    g1[2] = 1024u << 16;                  // tensor_dim1 = 1024 (bits 111:80)
    g1[3] = 256u << 16;                   // tile_dim0 = 256 (bits 127:112)
    g1[4] = 64u;                          // tile_dim1 = 64 rows
    g1[5] = 256u;                         // tensor_dim0_stride = 256
    g1[6] = 0u;
    g1[7] = 0u;
    asm volatile("tensor_load_to_lds %0, %1" :: "s"(g0), "s"(g1) : "memory");
    __builtin_amdgcn_s_wait_tensorcnt(0);
  }
  __syncthreads();

  const int m0 = blockIdx.x * 128 + wave * 16;
  int arow = m0 + lo;
  if (arow >= MTOT) arow = MTOT - 1;      // clamp (stores are guarded)
  const float* aptr = value + (size_t)arow * EMB + 2 * hi;

  v8f acc[4] = {};
#pragma unroll 4
  for (int k0 = 0; k0 < EMB; k0 += 4) {
    const v2f a = *(const v2f*)(aptr + k0);
#pragma unroll
    for (int tile = 0; tile < 4; ++tile) {
      const int c = tile * 16 + lo;
      const v2f bvec = *(const v2f*)(&ldsB[c * 260 + k0 + 2 * hi]);
      acc[tile] = wmma_f32(a, bvec, acc[tile]);
    }
  }

#pragma unroll
  for (int tile = 0; tile < 4; ++tile) {
    const int col = ncol0 + tile * 16 + lo;
    const float bias = b_val[col];
    const int h = col >> 5, d = col & 31;
#pragma unroll
    for (int i = 0; i < 8; ++i) {
      const int row = m0 + i + 8 * hi;    // D layout: vgpr i -> M = i + 8*hi
      if (row < MTOT) {
        const int bidx = row / NVAL;
        const int n = row - bidx * NVAL;
        vout[((size_t)(bidx * NH + h) * NVAL + n) * HD + d] = acc[tile][i] + bias;
      }
    }
  }
}

// ---------------------------------------------------------------------------
// Kernel 2 (fused): per (batch, 16-query tile):
//   off  = Q @ w_off  + b_off    (16x512 via WMMA, into LDS)
//   attn = softmax(Q @ w_attn + b_attn) over 32 per (q,head)   (LDS)
//   out  = sum_l sum_p attn * bilinear(v, ref + off/norm)
// ---------------------------------------------------------------------------
__global__ void __launch_bounds__(256)
msda_fused_kernel(const float* __restrict__ query, const float* __restrict__ refp,
                  const float* __restrict__ wT, const float* __restrict__ b_off,
                  const float* __restrict__ b_attn,
                  const float* __restrict__ vproj, float* __restrict__ out) {
  extern __shared__ float lds[];
  float* lq   = lds;            // 16 x 260 (padded: 260 mod 64 = 4 -> conflict-free)
  float* loff = lds + 4160;     // 16 x 512 raw offsets
  float* latt = lds + 12352;    // 16 x 256 attn logits -> weights
  float* lref = lds + 16448;    // 16 x 4 x 2 reference points

  const int t = threadIdx.x;
  const int wave = t >> 5, lane = t & 31, hi = lane >> 4, lo = lane & 15;
  const int b = blockIdx.y;
  const int q0 = blockIdx.x * 16;               // NQ % 16 == 0

  {
    const int i = t >> 4;
    const int cbase = (t & 15) * 4;
    const float* qrow = query + (size_t)(b * NQ + q0 + i) * EMB;
#pragma unroll
    for (int j = 0; j < 4; ++j) {
      const int c = cbase + j * 64;
      *(float4*)(&lq[i * 260 + c]) = *(const float4*)(qrow + c);
    }
  }
  if (t < 128) lref[t] = refp[(size_t)(b * NQ + q0) * 8 + t];
  __syncthreads();

  // 48 N-tiles over concatenated [off(512) | attn(256)]; B from transposed wT.
#pragma unroll 1
  for (int it = 0; it < 6; ++it) {
    const int nt = wave + it * 8;
    const int ncolFull = nt * 16;               // 0..767 in wT
    const float* bias; float* dst; int Nw, ncol;
    if (nt < 32) { bias = b_off;  dst = loff; Nw = 512; ncol = ncolFull; }
    else         { bias = b_attn; dst = latt; Nw = 256; ncol = ncolFull - 512; }
    v8f acc = {};
    const float* lqbase = lq + lo * 260 + 2 * hi;
    const float* wp = wT + (size_t)(ncolFull + lo) * EMB + 2 * hi;  // contiguous in k
#pragma unroll 4
    for (int k0 = 0; k0 < EMB; k0 += 4) {
      const v2f a = *(const v2f*)(lqbase + k0);
      const v2f bvec = *(const v2f*)(wp + k0);
      acc = wmma_f32(a, bvec, acc);
    }
    const float bv = bias[ncol + lo];
#pragma unroll
    for (int i = 0; i < 8; ++i) {
      const int qq = i + 8 * hi;
      dst[qq * Nw + ncol + lo] = acc[i] + bv;
    }
  }
  __syncthreads();

  // Softmax over the 32 (level,point) logits per (q, head).
  if (t < 128) {
    float* ap = latt + (t >> 3) * 256 + (t & 7) * 32;
    float m = ap[0];
#pragma unroll
    for (int j = 1; j < 32; ++j) m = fmaxf(m, ap[j]);
    float s = 0.f;
#pragma unroll
    for (int j = 0; j < 32; ++j) { const float e = expf(ap[j] - m); ap[j] = e; s += e; }
    const float inv = 1.f / s;
#pragma unroll
    for (int j = 0; j < 32; ++j) ap[j] *= inv;
  }
  __syncthreads();

  // Sampling: one wave per (q, head); lane = channel d (coalesced 128B reads).
  const int   WlT[4]  = {200, 100, 50, 25};
  const int   HlT[4]  = {116, 58, 29, 15};
  const int   SLT[4]  = {0, 23200, 29000, 30450};
  const float iWl[4]  = {1.f/200.f, 1.f/100.f, 1.f/50.f, 1.f/25.f};
  const float iHl[4]  = {1.f/116.f, 1.f/58.f, 1.f/29.f, 1.f/15.f};

#pragma unroll 1
  for (int pair = wave; pair < 128; pair += 8) {
    const int qq = pair >> 3, h = pair & 7;
    const float* attnRow = latt + qq * 256 + h * 32;
    const float* offRow  = loff + qq * 512 + h * 64;   // ((h*4+l)*8+p)*2
    const float* refRow  = lref + qq * 8;
    float acc = 0.f;
#pragma unroll
    for (int l = 0; l < 4; ++l) {
      const int Wi = WlT[l], Hi = HlT[l];
      const float fW = (float)Wi, fH = (float)Hi;
      const float* base = vproj + ((size_t)(b * NH + h) * NVAL + SLT[l]) * HD + lane;
#pragma unroll
      for (int p = 0; p < 8; ++p) {
        const float ox = offRow[l * 16 + p * 2];
        const float oy = offRow[l * 16 + p * 2 + 1];
        const int z = p & 3;                           // p -> (pz, z), z fastest
        const float x = (refRow[z * 2]     + ox * iWl[l]) * fW - 0.5f;
        const float y = (refRow[z * 2 + 1] + oy * iHl[l]) * fH - 0.5f;
        const float x0f = floorf(x), y0f = floorf(y);
        const float fx = x - x0f, fy = y - y0f;
        const int x0 = (int)x0f, y0 = (int)y0f;
        const int x1 = x0 + 1, y1 = y0 + 1;
        const bool x0v = (x0 >= 0) & (x0 < Wi);
        const bool x1v = (x1 >= 0) & (x1 < Wi);
        const bool y0v = (y0 >= 0) & (y0 < Hi);
        const bool y1v = (y1 >= 0) & (y1 < Hi);
        float s = 0.f;
        if (y0v) {
          const float* r0 = base + (size_t)y0 * Wi * HD;
          if (x0v) s += (1.f - fy) * (1.f - fx) * r0[(size_t)x0 * HD];
          if (x1v) s += (1.f - fy) * fx         * r0[(size_t)x1 * HD];
        }
        if (y1v) {
          const float* r1 = base + (size_t)y1 * Wi * HD;
          if (x0v) s += fy * (1.f - fx) * r1[(size_t)x0 * HD];
          if (x1v) s += fy * fx         * r1[(size_t)x1 * HD];
        }
        acc += attnRow[l * 8 + p] * s;
      }
    }
    out[(size_t)(b * NQ + q0 + qq) * EMB + h * HD + lane] = acc;
  }
}

extern "C" void kernel_launch(void* const* d_in, const int* in_sizes, int n_in,
                              void* d_out, int out_size, void* d_ws, size_t ws_size,
                              hipStream_t stream) {
  (void)in_sizes; (void)n_in; (void)out_size; (void)ws_size;
  const float* query  = (const float*)d_in[0];
  const float* value  = (const float*)d_in[1];
  const float* refp   = (const float*)d_in[2];
  const float* w_off  = (const float*)d_in[3];
  const float* b_off  = (const float*)d_in[4];
  const float* w_attn = (const float*)d_in[5];
  const float* b_attn = (const float*)d_in[6];
  const float* w_val  = (const float*)d_in[7];
  const float* b_val  = (const float*)d_in[8];
  float* out   = (float*)d_out;
  float* vproj = (float*)d_ws;                 // [6][8][30825][32] f32, 189.4 MB
  float* wT    = vproj + VPROJ_FLOATS;         // [1024][256] f32, 1 MB
  const float* wvT = wT + 768 * EMB;           // transposed w_val section

  wt_transpose_kernel<<<dim3(WT_COLS / 256), dim3(256), 0, stream>>>(
      w_off, w_attn, w_val, wT);

  dim3 g1((MTOT + 127) / 128, 4);
  val_proj_kernel<<<g1, dim3(256), 64 * 260 * sizeof(float), stream>>>(
      value, wvT, b_val, vproj);

  dim3 g2(NQ / 16, BS);
  msda_fused_kernel<<<g2, dim3(256), 16576 * sizeof(float), stream>>>(
      query, refp, wT, b_off, b_attn, vproj, out);
}